// ConstraintAndResult_75170517615175
// MI455X (gfx1250) — compile-verified
//
#include <hip/hip_runtime.h>

#define L 512
#define BATCH 64
#define BL (BATCH * L)
#define NTOT ((size_t)BATCH * L * L)
#define STEPS 20

typedef float v2f __attribute__((ext_vector_type(2)));
typedef float v8f __attribute__((ext_vector_type(8)));

// ---------------------------------------------------------------------------
// S_sym[b,i,j] = 0.5*(scores[b,i,j] + scores[b,j,i]) - s   (step-invariant)
// ---------------------------------------------------------------------------
__global__ void k_ssym(const float* __restrict__ sc, const float* __restrict__ s_p,
                       float* __restrict__ ssym) {
    __shared__ float T[32][33];
    const int b  = blockIdx.z;
    const int i0 = blockIdx.y * 32, j0 = blockIdx.x * 32;
    const int x = threadIdx.x, y = threadIdx.y;
    const float s = s_p[0];
    const float* base = sc + (size_t)b * L * L;
    float d[4];
#pragma unroll
    for (int k = 0; k < 4; k++) {
        int r = y + 8 * k;
        d[k]    = base[(i0 + r) * L + (j0 + x)];
        T[r][x] = base[(j0 + r) * L + (i0 + x)];
    }
    __syncthreads();
    float* out = ssym + (size_t)b * L * L;
#pragma unroll
    for (int k = 0; k < 4; k++) {
        int r = y + 8 * k;
        out[(i0 + r) * L + (j0 + x)] = 0.5f * (d[k] + T[x][r]) - s;
    }
}

// ---------------------------------------------------------------------------
// Strip kernel: each wave owns 16 rows x 256 cols of one batch slice.
// Tiles are held in the WMMA 16x16 f32 C-layout:
//   lane l, reg r  <->  (row = r + 8*(l>=16), col = l&15)
// INIT:  A_hat = scores * M  (mask folded in once; M symmetric => exact)
// STEP:  A_hat = min(relu(|A_hat*(1 + a_t*(Ssym - c_i - c_j))| - rho*a_t), 1)
// Column sums via matrix pipe: acc = ones(16x4) x W_chunk(4x16) + acc  (x4).
// Row sums via 4x shfl_xor (lane halves stay independent).
// ---------------------------------------------------------------------------
template <bool INIT>
__global__ void k_strip(const float* __restrict__ src0,   // INIT: scores
                        const float* __restrict__ Mmask,  // INIT: M
                        float* __restrict__ Ahat,
                        const float* __restrict__ ssym,
                        const float* __restrict__ rho,
                        const float* __restrict__ cvec,
                        float* __restrict__ rs, float* __restrict__ cs,
                        const float* __restrict__ alpha_p,
                        const float* __restrict__ lra_p, int t) {
    const int wid  = blockIdx.x * (blockDim.x >> 5) + (threadIdx.x >> 5);
    const int lane = threadIdx.x & 31;
    const int b    = wid >> 6;          // 64 strips per batch (32 row-bands x 2 col-halves)
    const int rem  = wid & 63;
    const int i0   = (rem >> 1) << 4;   // 16-row band
    const int jh   = rem & 1;           // which 256-col half
    const int hh   = lane >> 4;         // lane half -> +8 rows
    const int cl   = lane & 15;

    float a_t = 0.0f;
    if (!INIT) a_t = alpha_p[0] * __powf(lra_p[0], (float)t);

    const size_t bbase = (size_t)b * L * L;
    int   rowidx[8];
    float ci[8];
#pragma unroll
    for (int r = 0; r < 8; r++) {
        int row   = i0 + r + hh * 8;
        rowidx[r] = row * L;
        ci[r]     = INIT ? 0.0f : cvec[b * L + row];
    }
    float rowpart[8];
#pragma unroll
    for (int r = 0; r < 8; r++) rowpart[r] = 0.0f;

    v2f ones; ones.x = 1.0f; ones.y = 1.0f;

    for (int jt = 0; jt < 16; jt++) {
        const int j0  = jh * 256 + jt * 16;
        const int col = j0 + cl;
        const float cj = INIT ? 0.0f : cvec[b * L + col];
        float wt[8];
#pragma unroll
        for (int r = 0; r < 8; r++) {
            const size_t idx = bbase + (size_t)(rowidx[r] + col);
            float z;
            if (INIT) {
                z = src0[idx] * Mmask[idx];
            } else {
                float wv = Ahat[idx];
                float g  = ssym[idx] - ci[r] - cj;
                float x  = wv * fmaf(a_t, g, 1.0f);
                float y  = fmaxf(fabsf(x) - rho[rowidx[r] + col] * a_t, 0.0f);
                z = fminf(y, 1.0f);     // == 1 - relu(1 - y) for y >= 0
            }
            Ahat[idx]  = z;
            wt[r]      = z;
            rowpart[r] += z;
        }
        // column sums on the matrix pipe (K-row order is irrelevant for colsum)
        v8f acc = {0.f, 0.f, 0.f, 0.f, 0.f, 0.f, 0.f, 0.f};
#pragma unroll
        for (int q = 0; q < 4; q++) {
            v2f bv; bv.x = wt[2 * q]; bv.y = wt[2 * q + 1];
            acc = __builtin_amdgcn_wmma_f32_16x16x4_f32(
                false, ones, false, bv, (short)0, acc, false, false);
        }
        if (lane < 16) atomicAdd(&cs[b * L + j0 + lane], acc[0]);
    }
#pragma unroll
    for (int r = 0; r < 8; r++) {
        float v = rowpart[r];
        v += __shfl_xor(v, 1);
        v += __shfl_xor(v, 2);
        v += __shfl_xor(v, 4);
        v += __shfl_xor(v, 8);
        if (lane == 0)  atomicAdd(&rs[b * L + i0 + r], v);
        if (lane == 16) atomicAdd(&rs[b * L + i0 + r + 8], v);
    }
}

// ---------------------------------------------------------------------------
// Per-(b,i) vector update: row_dev, Lm, c = Lm*sign(row_dev); re-zero accums.
// ---------------------------------------------------------------------------
__global__ void k_vec(float* __restrict__ rs, float* __restrict__ cs,
                      float* __restrict__ Lm, float* __restrict__ cvec,
                      const float* __restrict__ w_p,
                      const float* __restrict__ belt_p,
                      const float* __restrict__ lrb_p, int t, int initMode) {
    const int i = blockIdx.x * blockDim.x + threadIdx.x;
    if (i >= BL) return;
    const float rd   = 0.5f * (rs[i] + cs[i]) - 1.0f;
    const float relu = fmaxf(rd, 0.0f);
    float lm;
    if (initMode) lm = w_p[0] * relu;
    else          lm = Lm[i] + belt_p[0] * __powf(lrb_p[0], (float)t) * relu;
    Lm[i] = lm;
    const float sg = (rd > 0.0f) ? 1.0f : ((rd < 0.0f) ? -1.0f : 0.0f);
    cvec[i] = lm * sg;
    rs[i] = 0.0f;
    cs[i] = 0.0f;
}

// ---------------------------------------------------------------------------
// Final in-place symmetrization: A = 0.5*(A_hat + A_hat^T)  (mask already in).
// Tile pairs (bi,bj)/(bj,bi), bj>=bi, owned by one block -> race-free.
// ---------------------------------------------------------------------------
__global__ void k_final(float* __restrict__ A) {
    const int bi = blockIdx.y, bj = blockIdx.x;
    if (bj < bi) return;
    __shared__ float T[32][33];
    __shared__ float U[32][33];
    const int b  = blockIdx.z;
    const int i0 = bi * 32, j0 = bj * 32;
    const int x = threadIdx.x, y = threadIdx.y;
    float* base = A + (size_t)b * L * L;
    float d[4];
#pragma unroll
    for (int k = 0; k < 4; k++) {
        int r = y + 8 * k;
        d[k]    = base[(i0 + r) * L + (j0 + x)];
        T[r][x] = base[(j0 + r) * L + (i0 + x)];
    }
    __syncthreads();
#pragma unroll
    for (int k = 0; k < 4; k++) {
        int r = y + 8 * k;
        float o = 0.5f * (d[k] + T[x][r]);
        base[(i0 + r) * L + (j0 + x)] = o;
        U[r][x] = o;
    }
    if (bi == bj) return;
    __syncthreads();
#pragma unroll
    for (int k = 0; k < 4; k++) {
        int r = y + 8 * k;
        base[(j0 + r) * L + (i0 + x)] = U[x][r];  // transposed twin tile
    }
}

extern "C" void kernel_launch(void* const* d_in, const int* in_sizes, int n_in,
                              void* d_out, int out_size, void* d_ws, size_t ws_size,
                              hipStream_t stream) {
    const float* scores = (const float*)d_in[0];
    const float* M      = (const float*)d_in[1];
    const float* s_p    = (const float*)d_in[2];
    const float* w_p    = (const float*)d_in[3];
    const float* rho    = (const float*)d_in[4];
    const float* alpha  = (const float*)d_in[5];
    const float* belt   = (const float*)d_in[6];
    const float* lra    = (const float*)d_in[7];
    const float* lrb    = (const float*)d_in[8];

    float* Ahat = (float*)d_out;          // A_hat lives in d_out during the loop
    float* ws   = (float*)d_ws;
    float* ssym = ws;                     // [B,L,L]  64 MB
    float* rs   = ws + NTOT;              // [B,L]
    float* cs   = rs + BL;                // [B,L]
    float* Lm   = cs + BL;                // [B,L]
    float* cv   = Lm + BL;                // [B,L]

    hipMemsetAsync(rs, 0, (size_t)2 * BL * sizeof(float), stream);

    k_ssym<<<dim3(16, 16, 64), dim3(32, 8), 0, stream>>>(scores, s_p, ssym);

    k_strip<true><<<1024, 128, 0, stream>>>(scores, M, Ahat, nullptr, nullptr,
                                            nullptr, rs, cs, nullptr, nullptr, 0);
    k_vec<<<BL / 256, 256, 0, stream>>>(rs, cs, Lm, cv, w_p, belt, lrb, 0, 1);

    for (int t = 0; t < STEPS; t++) {
        k_strip<false><<<1024, 128, 0, stream>>>(nullptr, nullptr, Ahat, ssym,
                                                 rho, cv, rs, cs, alpha, lra, t);
        if (t < STEPS - 1)
            k_vec<<<BL / 256, 256, 0, stream>>>(rs, cs, Lm, cv, w_p, belt, lrb, t, 0);
    }

    k_final<<<dim3(16, 16, 64), dim3(32, 8), 0, stream>>>(Ahat);
}